// LossYOLO_v2_8761733284305
// MI455X (gfx1250) — compile-verified
//
#include <hip/hip_runtime.h>
#include <math.h>

// ---------------- problem constants ----------------
#define NUM_CLASSES 20
#define NUM_ANC     5
#define S_          21          // 1 + NUM_CLASSES
#define GRID        38
#define HW          1444        // 38*38
#define NANCH       7220        // HW * NUM_ANC
#define BATCH       64
#define NBOX        16
#define CPLANES     125         // (S_+4)*NUM_ANC
#define BSTRIDE     180500      // CPLANES*HW
#define EPS16F      0.0009765625f

// workspace layout (floats): accum[5][64] banked partial sums
#define ACC_OFF 0
#define WS_FLOATS (5*64)

__device__ __constant__ float d_AW[NUM_ANC] = {0.05f, 0.12f, 0.25f, 0.45f, 0.75f};
__device__ __constant__ float d_AH[NUM_ANC] = {0.07f, 0.15f, 0.30f, 0.55f, 0.80f};

typedef __attribute__((ext_vector_type(2))) float v2f;
typedef __attribute__((ext_vector_type(8))) float v8f;

// sigmoid followed by the x_bce / pconf clip (reference clips to [eps, 1-eps])
__device__ inline float sigclip(float x) {
    float s = 1.0f / (1.0f + expf(-x));
    return fminf(fmaxf(s, EPS16F), 1.0f - EPS16F);
}

// Reproduce match_one per box: returns cell, chosen anchor, and regression targets.
__device__ inline void match_box(float l, float t, float r, float bt,
                                 int* cell, int* anc,
                                 float* tx, float* ty, float* tw, float* th) {
    float cx = (l + r) * 0.5f, cy = (t + bt) * 0.5f;
    float w = r - l, h = bt - t;
    int best = 0; float bestiou = -1e30f;
    #pragma unroll
    for (int a = 0; a < NUM_ANC; ++a) {
        float inter = fminf(w, d_AW[a]) * fminf(h, d_AH[a]);
        float uni   = w * h + d_AW[a] * d_AH[a] - inter;
        float iou   = inter / uni;
        if (iou > bestiou) { bestiou = iou; best = a; }   // jnp.argmax: first max wins
    }
    int col = (int)floorf(cx * (float)GRID); col = min(max(col, 0), GRID - 1);
    int row = (int)floorf(cy * (float)GRID); row = min(max(row, 0), GRID - 1);
    *cell = row * GRID + col;
    *anc  = best;
    *tx = cx * (float)GRID - (float)col;
    *ty = cy * (float)GRID - (float)row;
    *tw = logf(w / d_AW[best]);
    *th = logf(h / d_AH[best]);
}

// exact 64-value f32 sum across one wave using V_WMMA_F32_16X16X4_F32
// (A = ones(16x4) -> D[m][n] = sum over the 4 B values feeding column n;
//  columns n and n+16 lane-groups both carry colsum[n], hence the *0.5)
__device__ inline float reduce64_wmma(const float* p, int lane) {
    float b0 = p[lane], b1 = p[lane + 32];
#if defined(__gfx1250__)
    v2f a;  a[0] = 1.0f; a[1] = 1.0f;
    v2f bv; bv[0] = b0;  bv[1] = b1;
    v8f c = {};
    v8f d = __builtin_amdgcn_wmma_f32_16x16x4_f32(false, a, false, bv,
                                                  (short)0, c, false, false);
    float s = d[0];
    #pragma unroll
    for (int o = 16; o > 0; o >>= 1) s += __shfl_xor(s, o, 32);
    return 0.5f * s;
#else
    float s = b0 + b1;
    #pragma unroll
    for (int o = 16; o > 0; o >>= 1) s += __shfl_xor(s, o, 32);
    return s;
#endif
}

// ---------------- kernel 1: zero the banked accumulators ----------------
__global__ void init_kernel(float* wsf) {
    int i = threadIdx.x;                 // 320 threads
    if (i < WS_FLOATS) wsf[ACC_OFF + i] = 0.0f;
}

// ---------------- kernel 2: main streaming loss ----------------
// grid = (ceil(HW/256), BATCH), block = 256
__global__ void loss_kernel(const float* __restrict__ py,
                            const float* __restrict__ gb,
                            const int*   __restrict__ gl,
                            float* wsf) {
    __shared__ float sent[NBOX * 8];     // per-box: cell, anc, label, tx, ty, tw, th, wpos
    __shared__ float wpart[8][5];

    const int b   = blockIdx.y;
    const int tid = threadIdx.x;

    // threads 0..15: recompute the (trivial) matching for this batch into LDS
    if (tid < NBOX) {
        const float* G = gb + (size_t)b * NBOX * 4 + tid * 4;
        int cell, anc; float tx, ty, tw, th;
        match_box(G[0], G[1], G[2], G[3], &cell, &anc, &tx, &ty, &tw, &th);
        sent[tid * 8 + 0] = (float)cell;
        sent[tid * 8 + 1] = (float)anc;
        sent[tid * 8 + 2] = (float)(gl[(size_t)b * NBOX + tid] - 1);
        sent[tid * 8 + 3] = tx;
        sent[tid * 8 + 4] = ty;
        sent[tid * 8 + 5] = tw;
        sent[tid * 8 + 6] = th;
        sent[tid * 8 + 7] = G[0];        // reference quirk: w_pos = gy[...,26] = left coord
    }
    __syncthreads();

    const int s = blockIdx.x * blockDim.x + tid;
    float acc[5] = {0.f, 0.f, 0.f, 0.f, 0.f};  // 0:l_pos 1:l_neg 2:l_cls 3:l_txty 4:l_twth

    if (s < HW) {
        const float* base = py + (size_t)b * BSTRIDE;
        bool cellmatch = false;
        #pragma unroll
        for (int i = 0; i < NBOX; ++i)
            if ((int)sent[i * 8] == s) cellmatch = true;

        #pragma unroll
        for (int a = 0; a < NUM_ANC; ++a) {
            // conf channel: c = 0*5 + a  (only bytes the hot path touches)
            float pc = sigclip(base[a * HW + s]);
            if (!cellmatch) {
                // gconf==0 -> mask_neg; l_neg = -0.5*p^2*log(1-p)
                acc[1] += -0.5f * pc * pc * logf(1.0f - pc);
            } else {
                // scatter semantics: pass1 set -1 on all anchors of matched cells,
                // pass2 last-box-wins on the chosen anchor
                int hit = -1;
                for (int i = 0; i < NBOX; ++i)
                    if ((int)sent[i * 8] == s && (int)sent[i * 8 + 1] == a) hit = i;
                if (hit >= 0) {
                    // positive anchor (gconf == 1)
                    acc[0] += -0.5f * (1.0f - pc) * (1.0f - pc) * logf(pc);
                    const int lab = (int)sent[hit * 8 + 2];
                    float cls = 0.f;
                    #pragma unroll
                    for (int ch = 1; ch <= NUM_CLASSES; ++ch) {
                        float p = sigclip(base[(ch * NUM_ANC + a) * HW + s]);
                        cls += (ch - 1 == lab) ? -logf(p) : -logf(1.0f - p);
                    }
                    acc[2] += cls;
                    const float wp = sent[hit * 8 + 7];
                    const float tx = sent[hit * 8 + 3], ty = sent[hit * 8 + 4];
                    float px  = sigclip(base[(21 * NUM_ANC + a) * HW + s]);
                    float pyv = sigclip(base[(22 * NUM_ANC + a) * HW + s]);
                    float bce = -(tx * logf(px)  + (1.0f - tx) * logf(1.0f - px))
                              - (ty * logf(pyv) + (1.0f - ty) * logf(1.0f - pyv));
                    acc[3] += bce * wp;
                    const float tw = sent[hit * 8 + 5], th = sent[hit * 8 + 6];
                    float pw = base[(23 * NUM_ANC + a) * HW + s];
                    float ph = base[(24 * NUM_ANC + a) * HW + s];
                    acc[4] += ((pw - tw) * (pw - tw) + (ph - th) * (ph - th)) * wp;
                }
                // hit < 0: conf == -1 -> excluded from both masks
            }
        }
    }

    // wave32 butterfly reduce, then cross-wave via LDS, one atomic per quantity per block
    #pragma unroll
    for (int q = 0; q < 5; ++q)
        for (int o = 16; o > 0; o >>= 1) acc[q] += __shfl_xor(acc[q], o, 32);
    const int wave = tid >> 5;
    if ((tid & 31) == 0)
        for (int q = 0; q < 5; ++q) wpart[wave][q] = acc[q];
    __syncthreads();
    if (tid == 0) {
        const int bank = (blockIdx.y * gridDim.x + blockIdx.x) & 63;
        for (int q = 0; q < 5; ++q) {
            float v = 0.f;
            for (int w2 = 0; w2 < 8; ++w2) v += wpart[w2][q];
            atomicAdd(&wsf[ACC_OFF + q * 64 + bank], v);
        }
    }
}

// ---------------- kernel 3: finalize (one wave32) ----------------
__global__ void finalize_kernel(const float* __restrict__ wsf,
                                const float* __restrict__ gb,
                                float* out) {
    const int lane = threadIdx.x;        // exactly 32 threads -> EXEC all ones for WMMA

    // exact WMMA-based 64->1 reduction of each banked accumulator
    float tot[5];
    for (int q = 0; q < 5; ++q)
        tot[q] = reduce64_wmma(wsf + ACC_OFF + q * 64, lane);

    // per-batch pos/neg counts: each lane handles 2 batches (O(16^2) dedup)
    float invp = 0.f, invn = 0.f, npt = 0.f;
    for (int k = 0; k < 2; ++k) {
        const int b = lane + k * 32;
        const float* G = gb + (size_t)b * NBOX * 4;
        int cells[NBOX], ancs[NBOX];
        for (int i = 0; i < NBOX; ++i) {
            float tx, ty, tw, th;
            match_box(G[i * 4 + 0], G[i * 4 + 1], G[i * 4 + 2], G[i * 4 + 3],
                      &cells[i], &ancs[i], &tx, &ty, &tw, &th);
        }
        int D = 0, P = 0;
        for (int i = 0; i < NBOX; ++i) {
            bool firstCell = true, firstPair = true;
            for (int j = 0; j < i; ++j) {
                if (cells[j] == cells[i]) {
                    firstCell = false;
                    if (ancs[j] == ancs[i]) firstPair = false;
                }
            }
            if (firstCell) ++D;
            if (firstPair) ++P;
        }
        float np = (float)P;
        float nn = (float)(NANCH - 5 * D);
        npt  += np;
        invp += 1.0f / fmaxf(np, EPS16F);
        invn += 1.0f / fmaxf(nn, EPS16F);
    }
    #pragma unroll
    for (int o = 16; o > 0; o >>= 1) {
        invp += __shfl_xor(invp, o, 32);
        invn += __shfl_xor(invn, o, 32);
        npt  += __shfl_xor(npt,  o, 32);
    }

    if (lane == 0) {
        invp *= (1.0f / (float)BATCH);   // mean(1/nums_pos)
        invn *= (1.0f / (float)BATCH);   // mean(1/nums_neg)
        float n_pos_tot = fmaxf(npt, 1.0f);
        float loss = tot[0] * invp                 // l_conf_pos
                   + 3.0f * tot[1] * invn          // l_conf_neg
                   + tot[2] / n_pos_tot            // l_cls (LOSS_W_CLS = 1)
                   + tot[3] / n_pos_tot            // l_txty
                   + tot[4] / n_pos_tot;           // l_twth
        out[0] = loss;
    }
}

// ---------------- host launcher ----------------
extern "C" void kernel_launch(void* const* d_in, const int* in_sizes, int n_in,
                              void* d_out, int out_size, void* d_ws, size_t ws_size,
                              hipStream_t stream) {
    (void)in_sizes; (void)n_in; (void)out_size; (void)ws_size;
    const float* py = (const float*)d_in[0];   // pyolos  (64,125,38,38) f32
    const float* gb = (const float*)d_in[1];   // gboxes_ltrb (64,16,4)  f32
    const int*   gl = (const int*)d_in[2];     // glabels (64,16)        i32
    float* wsf = (float*)d_ws;
    float* out = (float*)d_out;

    init_kernel<<<1, WS_FLOATS, 0, stream>>>(wsf);
    dim3 grid((HW + 255) / 256, BATCH);
    loss_kernel<<<grid, 256, 0, stream>>>(py, gb, gl, wsf);
    finalize_kernel<<<1, 32, 0, stream>>>(wsf, gb, out);
}